// GNNNetwork_81905026335010
// MI455X (gfx1250) — compile-verified
//
#include <hip/hip_runtime.h>
#include <math.h>

typedef __attribute__((ext_vector_type(2))) float v2f;
typedef __attribute__((ext_vector_type(8))) float v8f;

#define H64 64

// ---------------------------------------------------------------------------
// Degree count: deg[dst] += 1  (float atomics; deg zeroed by memset)
__global__ void k_count_deg(const long long* __restrict__ dst, float* __restrict__ deg, int E) {
    int e = blockIdx.x * blockDim.x + threadIdx.x;
    if (e < E) atomicAdd(&deg[(int)dst[e]], 1.0f);
}

// dinv[i] = rsqrt(deg[i] + 1)   (+1 self loop)
__global__ void k_dinv(const float* __restrict__ deg, float* __restrict__ dinv, int N) {
    int i = blockIdx.x * blockDim.x + threadIdx.x;
    if (i < N) dinv[i] = rsqrtf(deg[i] + 1.0f);
}

// Pad Wl [128,40] -> Wp [128,48] with zero columns
__global__ void k_pad_wl(const float* __restrict__ Wl, float* __restrict__ Wp) {
    int idx = blockIdx.x * blockDim.x + threadIdx.x;
    if (idx >= 128 * 48) return;
    int k = idx / 48, n = idx % 48;
    Wp[idx] = (n < 40) ? Wl[k * 40 + n] : 0.0f;
}

// ---------------------------------------------------------------------------
// Generic fp32 WMMA GEMM: C[M,Nn] = A[M,K] * B[K,Nn]
// One wave computes one 16x16 tile; wave w in block -> n-tile w; block -> m-tile.
// Requires: K % 4 == 0, Nn % 16 == 0, blockDim.x == 32 * (Nn/16).
__global__ void k_gemm_wmma(const float* __restrict__ A, const float* __restrict__ B,
                            float* __restrict__ C, int M, int K, int Nn) {
    const int lane = threadIdx.x & 31;
    const int wave = threadIdx.x >> 5;
    const int m0 = blockIdx.x * 16;
    const int n0 = wave * 16;
    const int half = lane >> 4;   // 0: K pair {0,1}, 1: K pair {2,3}
    const int l16  = lane & 15;

    int row = m0 + l16;
    if (row >= M) row = M - 1;               // clamp: keep EXEC all-1s for WMMA
    const float* arow = A + (size_t)row * K;

    v8f acc = {};
    for (int k = 0; k < K; k += 4) {
        const int ka = k + 2 * half;
        v2f a, b;
        a.x = arow[ka];
        a.y = arow[ka + 1];
        b.x = B[(size_t)ka * Nn + n0 + l16];
        b.y = B[(size_t)(ka + 1) * Nn + n0 + l16];
        acc = __builtin_amdgcn_wmma_f32_16x16x4_f32(false, a, false, b,
                                                    (short)0, acc, false, false);
    }
#pragma unroll
    for (int r = 0; r < 8; ++r) {
        int m = m0 + r + 8 * half;
        if (m < M) C[(size_t)m * Nn + n0 + l16] = acc[r];
    }
}

// Concat GEMM + bias: logits[M,40] = [X1|X2][M,128] * Wp[128,48] + bl
// blockDim.x = 96 (3 waves -> 3 n-tiles covering 48 padded cols)
__global__ void k_gemm_concat(const float* __restrict__ X1, const float* __restrict__ X2,
                              const float* __restrict__ Wp, const float* __restrict__ bl,
                              float* __restrict__ Cout, int M) {
    const int lane = threadIdx.x & 31;
    const int wave = threadIdx.x >> 5;
    const int m0 = blockIdx.x * 16;
    const int n0 = wave * 16;
    const int half = lane >> 4;
    const int l16  = lane & 15;

    int row = m0 + l16;
    if (row >= M) row = M - 1;
    const float* a1 = X1 + (size_t)row * H64;
    const float* a2 = X2 + (size_t)row * H64;

    v8f acc = {};
#pragma unroll
    for (int k = 0; k < 128; k += 4) {
        const int ka = k + 2 * half;
        const float* ap = (k < 64) ? (a1 + ka) : (a2 + (ka - 64));
        v2f a, b;
        a.x = ap[0];
        a.y = ap[1];
        b.x = Wp[(size_t)ka * 48 + n0 + l16];
        b.y = Wp[(size_t)(ka + 1) * 48 + n0 + l16];
        acc = __builtin_amdgcn_wmma_f32_16x16x4_f32(false, a, false, b,
                                                    (short)0, acc, false, false);
    }
    const int col = n0 + l16;
    if (col < 40) {
        const float bv = bl[col];
#pragma unroll
        for (int r = 0; r < 8; ++r) {
            int m = m0 + r + 8 * half;
            if (m < M) Cout[(size_t)m * 40 + col] = acc[r] + bv;
        }
    }
}

// ---------------------------------------------------------------------------
// Edge aggregation: agg[dst] += h[src] * (dinv[src]*dinv[dst])
// 32 lanes per edge, each lane owns a float2 of the 64-wide feature row.
__global__ void k_edge_agg(const long long* __restrict__ src, const long long* __restrict__ dst,
                           const float* __restrict__ dinv, const float* __restrict__ h,
                           float* __restrict__ agg, int E) {
    long long tid = (long long)blockIdx.x * blockDim.x + threadIdx.x;
    int e = (int)(tid >> 5);
    if (e >= E) return;
    int lane = (int)(tid & 31);
    int s = (int)src[e];
    int d = (int)dst[e];
    float nrm = dinv[s] * dinv[d];
    const float2* hs = (const float2*)(h + (size_t)s * H64);
    float2 v = hs[lane];
    float* ad = agg + (size_t)d * H64 + lane * 2;
    atomicAdd(ad,     v.x * nrm);
    atomicAdd(ad + 1, v.y * nrm);
}

// Self-loop + bias: x[i,c] += h[i,c] * dinv[i]^2 + b[c]   (in place on agg)
__global__ void k_selfloop_bias(float* __restrict__ x, const float* __restrict__ h,
                                const float* __restrict__ dinv, const float* __restrict__ b,
                                long long total) {
    long long idx = (long long)blockIdx.x * blockDim.x + threadIdx.x;
    if (idx >= total) return;
    int i = (int)(idx >> 6);
    int c = (int)(idx & 63);
    float di = dinv[i];
    x[idx] += h[idx] * di * di + b[c];
}

// BatchNorm stats: stats[c] = sum_i x[i,c], stats[64+c] = sum_i x[i,c]^2
__global__ void k_bn_stats(const float* __restrict__ x, float* __restrict__ stats, int N) {
    const int c = threadIdx.x & 63;
    const int r = threadIdx.x >> 6;       // 0..3 (blockDim = 256)
    float s = 0.f, ss = 0.f;
    for (int i = blockIdx.x * 4 + r; i < N; i += gridDim.x * 4) {
        float v = x[(size_t)i * H64 + c];
        s += v;
        ss += v * v;
    }
    __shared__ float shs[4][64];
    __shared__ float shq[4][64];
    shs[r][c] = s;
    shq[r][c] = ss;
    __syncthreads();
    if (r == 0) {
        s  = shs[0][c] + shs[1][c] + shs[2][c] + shs[3][c];
        ss = shq[0][c] + shq[1][c] + shq[2][c] + shq[3][c];
        atomicAdd(&stats[c], s);
        atomicAdd(&stats[64 + c], ss);
    }
}

// BN (training-mode batch stats) + ReLU, in place
__global__ void k_bn_relu(float* __restrict__ x, const float* __restrict__ stats,
                          const float* __restrict__ g, const float* __restrict__ be,
                          int N, long long total) {
    long long idx = (long long)blockIdx.x * blockDim.x + threadIdx.x;
    if (idx >= total) return;
    int c = (int)(idx & 63);
    float invN = 1.0f / (float)N;
    float mu = stats[c] * invN;
    float var = stats[64 + c] * invN - mu * mu;
    float v = (x[idx] - mu) * rsqrtf(var + 1e-5f) * g[c] + be[c];
    x[idx] = v > 0.f ? v : 0.f;
}

// Row-wise log_softmax over 40 logits; one wave per row.
__global__ void k_log_softmax40(const float* __restrict__ logits, float* __restrict__ out, int N) {
    int row  = blockIdx.x * (blockDim.x >> 5) + (threadIdx.x >> 5);
    int lane = threadIdx.x & 31;
    if (row >= N) return;
    const float* p = logits + (size_t)row * 40;
    float v0 = p[lane];                                        // lanes 0..31 (<40)
    float v1 = (lane < 8) ? p[32 + lane] : -INFINITY;
    float m = fmaxf(v0, v1);
#pragma unroll
    for (int o = 16; o > 0; o >>= 1) m = fmaxf(m, __shfl_xor(m, o, 32));
    float s = __expf(v0 - m) + ((lane < 8) ? __expf(v1 - m) : 0.f);
#pragma unroll
    for (int o = 16; o > 0; o >>= 1) s += __shfl_xor(s, o, 32);
    float lse = m + __logf(s);
    float* q = out + (size_t)row * 40;
    q[lane] = v0 - lse;
    if (lane < 8) q[32 + lane] = v1 - lse;
}

// ---------------------------------------------------------------------------
extern "C" void kernel_launch(void* const* d_in, const int* in_sizes, int n_in,
                              void* d_out, int out_size, void* d_ws, size_t ws_size,
                              hipStream_t stream) {
    const float*     x0  = (const float*)d_in[0];
    const long long* ei  = (const long long*)d_in[1];   // int64 [2, E]
    const float*     W1  = (const float*)d_in[2];
    const float*     b1  = (const float*)d_in[3];
    const float*     g1  = (const float*)d_in[4];
    const float*     be1 = (const float*)d_in[5];
    const float*     W2  = (const float*)d_in[6];
    const float*     b2  = (const float*)d_in[7];
    const float*     Wl  = (const float*)d_in[8];
    const float*     bl  = (const float*)d_in[9];
    float* out = (float*)d_out;

    const int N = in_sizes[0] / 128;     // 100000
    const int E = in_sizes[1] / 2;       // 1600000
    const long long NH = (long long)N * H64;

    // workspace layout (floats)
    float* ws     = (float*)d_ws;
    float* deg    = ws;                         // N
    float* dinv   = deg    + N;                 // N
    float* h      = dinv   + N;                 // N*64  (h1, then h2)
    float* pre1   = h      + NH;                // N*64  (agg1 -> pre1 -> x1)
    float* x2b    = pre1   + NH;                // N*64  (agg2 -> x2)
    float* logits = x2b    + NH;                // N*40
    float* stats  = logits + (long long)N * 40; // 128
    float* Wp     = stats  + 128;               // 128*48

    const long long* srcI = ei;
    const long long* dstI = ei + E;

    // zero the accumulators (graph-capturable memset nodes)
    hipMemsetAsync(deg,  0, (size_t)N * sizeof(float), stream);
    hipMemsetAsync(pre1, 0, (size_t)NH * sizeof(float), stream);
    hipMemsetAsync(x2b,  0, (size_t)NH * sizeof(float), stream);
    hipMemsetAsync(stats, 0, 128 * sizeof(float), stream);

    // degree / normalization
    k_count_deg<<<(E + 255) / 256, 256, 0, stream>>>(dstI, deg, E);
    k_dinv<<<(N + 255) / 256, 256, 0, stream>>>(deg, dinv, N);
    k_pad_wl<<<(128 * 48 + 255) / 256, 256, 0, stream>>>(Wl, Wp);

    const int mtiles = (N + 15) / 16;   // 6250

    // layer 1: h1 = x0 @ W1
    k_gemm_wmma<<<mtiles, 128, 0, stream>>>(x0, W1, h, N, 128, 64);
    // agg1[dst] += h1[src]*norm
    {
        long long thr = (long long)E * 32;
        k_edge_agg<<<(unsigned)((thr + 255) / 256), 256, 0, stream>>>(srcI, dstI, dinv, h, pre1, E);
    }
    // pre1 += h1 * dinv^2 + b1
    k_selfloop_bias<<<(unsigned)((NH + 255) / 256), 256, 0, stream>>>(pre1, h, dinv, b1, NH);
    // batchnorm + relu -> x1 (in place)
    k_bn_stats<<<256, 256, 0, stream>>>(pre1, stats, N);
    k_bn_relu<<<(unsigned)((NH + 255) / 256), 256, 0, stream>>>(pre1, stats, g1, be1, N, NH);

    // layer 2: h2 = x1 @ W2
    k_gemm_wmma<<<mtiles, 128, 0, stream>>>(pre1, W2, h, N, 64, 64);
    {
        long long thr = (long long)E * 32;
        k_edge_agg<<<(unsigned)((thr + 255) / 256), 256, 0, stream>>>(srcI, dstI, dinv, h, x2b, E);
    }
    k_selfloop_bias<<<(unsigned)((NH + 255) / 256), 256, 0, stream>>>(x2b, h, dinv, b2, NH);

    // final linear on concat + bias
    k_gemm_concat<<<mtiles, 96, 0, stream>>>(pre1, x2b, Wp, bl, logits, N);

    // log_softmax (8 rows per 256-thread block)
    k_log_softmax40<<<(N + 7) / 8, 256, 0, stream>>>(logits, out, N);
}